// GroupMLP_28200755265770
// MI455X (gfx1250) — compile-verified
//
#include <hip/hip_runtime.h>
#include <hip/hip_bf16.h>

// GroupMLP for MI455X (gfx1250): per-head GEMM -> batch-BN -> ReLU -> per-head GEMM.
// fp32 WMMA (V_WMMA_F32_16X16X4_F32) for full precision. b1 cancels under
// training-mode BN (never read). Weights are repacked once per launch into
// WMMA-fragment order so every B fragment is one coalesced global_load_b64
// with an immediate offset. A fragments are single ds_load_b64 from a
// 64 KB XOR-swizzled LDS tile.

typedef float v2f __attribute__((ext_vector_type(2)));
typedef float v8f __attribute__((ext_vector_type(8)));

#define HEADS    8
#define FIN      128
#define HID      128
#define FOUT     64
#define ROWS     128     // rows (batch elements) per block
#define NTHREADS 256     // 8 wave32 waves
#define KSTEPS   32      // 128 / 4

// XOR swizzle for the 128x128 fp32 LDS tile (exactly 64 KB):
// bank(elem) = (c ^ 4*(r&15)) & 63 -> conflict-free for 16 lanes reading the
// same column pair across consecutive rows; preserves 16B groups for float4
// staging and 8B pairs for float2 fragment loads.
__device__ __forceinline__ int sw_idx(int r, int c) {
    return r * 128 + (c ^ ((r & 15) << 2));
}

__device__ __forceinline__ v8f wmma_f32(v2f a, v2f b, v8f c) {
    return __builtin_amdgcn_wmma_f32_16x16x4_f32(false, a, false, b,
                                                 (short)0, c, false, false);
}

// ---------------------------------------------------------------------------
// Weight repack into WMMA-fragment order:
//   pw1[((h*32 + k)*8 + t)*64 + L*2 + c] = w1[h][4k + c + 2*(L>=16)][t*16 + (L&15)]
//   pw2[((h*32 + k)*4 + t)*64 + L*2 + c] = w2[h][4k + c + 2*(L>=16)][t*16 + (L&15)]
// ---------------------------------------------------------------------------
__global__ __launch_bounds__(NTHREADS)
void pack_weights_kernel(const float* __restrict__ w1,
                         const float* __restrict__ w2,
                         float* __restrict__ pw1,
                         float* __restrict__ pw2)
{
    const int idx = blockIdx.x * NTHREADS + threadIdx.x;
    if (idx < HEADS * KSTEPS * 8 * 64) {                 // 131072 (512 KB)
        const int c = idx & 1;
        const int L = (idx >> 1) & 31;
        const int t = (idx >> 6) & 7;
        const int k = (idx >> 9) & 31;
        const int h = idx >> 14;
        const int kk = 4 * k + c + ((L >= 16) ? 2 : 0);
        pw1[idx] = w1[((size_t)h * FIN + kk) * HID + t * 16 + (L & 15)];
    }
    if (idx < HEADS * KSTEPS * 4 * 64) {                 // 65536 (256 KB)
        const int c = idx & 1;
        const int L = (idx >> 1) & 31;
        const int t = (idx >> 6) & 3;
        const int k = (idx >> 8) & 31;
        const int h = idx >> 13;
        const int kk = 4 * k + c + ((L >= 16) ? 2 : 0);
        pw2[idx] = w2[((size_t)h * HID + kk) * FOUT + t * 16 + (L & 15)];
    }
}

// Stage one 128x128 x-tile into swizzled LDS (zero-pad invalid rows).
__device__ __forceinline__ void stage_x(const float* __restrict__ x,
                                        float* __restrict__ xs,
                                        int row0, int head, int N, int tid)
{
    for (int i = tid; i < ROWS * (FIN / 4); i += NTHREADS) {
        const int r  = i >> 5;
        const int c4 = (i & 31) << 2;
        float4 v = make_float4(0.f, 0.f, 0.f, 0.f);
        if (row0 + r < N)
            v = *(const float4*)(x + (size_t)(row0 + r) * (HEADS * FIN)
                                   + head * FIN + c4);
        *(float4*)&xs[sw_idx(r, c4)] = v;
    }
}

// GEMM1 inner product: acc[8] += xs[16x128 tile rows rA..] @ pw1 fragments.
// arow already includes rA*128 + koff; sA = 4*(rA&15); (4k+koff)^sA = (4k^sA)+koff.
__device__ __forceinline__ void gemm1_core(const float* __restrict__ arow,
                                           const float* __restrict__ pw1hl,
                                           int sA, v8f acc[8])
{
    #pragma unroll
    for (int k = 0; k < KSTEPS; ++k) {
        const v2f a = *(const v2f*)(arow + ((4 * k) ^ sA));
        #pragma unroll
        for (int t = 0; t < 8; ++t) {
            const v2f b = *(const v2f*)(pw1hl + (k * 8 + t) * 64);
            acc[t] = wmma_f32(a, b, acc[t]);
        }
    }
}

// ---------------------------------------------------------------------------
// Kernel 1: mm = x @ w1 (no bias), accumulate per-(head,channel) moments.
// Zero-padded tail rows contribute exactly 0 to both moments.
// ---------------------------------------------------------------------------
__global__ __launch_bounds__(NTHREADS)
void gemm1_stats_kernel(const float* __restrict__ x,
                        const float* __restrict__ pw1,
                        float* __restrict__ gsum,
                        float* __restrict__ gsumsq,
                        int N)
{
    __shared__ float xs[ROWS * 128];             // 64 KB, swizzled

    const int head = blockIdx.y;
    const int row0 = blockIdx.x * ROWS;
    const int tid  = threadIdx.x;

    stage_x(x, xs, row0, head, N, tid);
    __syncthreads();

    const int wave = tid >> 5;
    const int lane = tid & 31;
    const int m    = lane & 15;
    const int koff = (lane < 16) ? 0 : 2;
    const int rA   = (wave << 4) + m;
    const int sA   = (rA & 15) << 2;

    const float* arow  = xs + rA * 128 + koff;
    const float* pw1hl = pw1 + (size_t)head * (KSTEPS * 8 * 64) + lane * 2;

    v8f acc[8];
    #pragma unroll
    for (int t = 0; t < 8; ++t)
        #pragma unroll
        for (int j = 0; j < 8; ++j) acc[t][j] = 0.f;

    gemm1_core(arow, pw1hl, sA, acc);

    // Moments: D layout is (M = j + 8*(lane>=16), N = t*16 + m).
    #pragma unroll
    for (int t = 0; t < 8; ++t) {
        float s = 0.f, q = 0.f;
        #pragma unroll
        for (int j = 0; j < 8; ++j) { float v = acc[t][j]; s += v; q += v * v; }
        s += __shfl_xor(s, 16, 32);              // fold lane halves (wave32)
        q += __shfl_xor(q, 16, 32);
        if (lane < 16) {
            atomicAdd(&gsum  [head * HID + t * 16 + lane], s);
            atomicAdd(&gsumsq[head * HID + t * 16 + lane], q);
        }
    }
}

// ---------------------------------------------------------------------------
// Kernel 2: scale = bn_w * rsqrt(var + eps); shift = bn_b - mean * scale.
// ---------------------------------------------------------------------------
__global__ __launch_bounds__(128)
void finalize_kernel(const float* __restrict__ gsum,
                     const float* __restrict__ gsumsq,
                     const float* __restrict__ bn_w,
                     const float* __restrict__ bn_b,
                     float* __restrict__ scale,
                     float* __restrict__ shift,
                     int N)
{
    const int idx  = blockIdx.x * 128 + threadIdx.x;
    const float invN = 1.0f / (float)N;
    const float mean = gsum[idx] * invN;
    float var = gsumsq[idx] * invN - mean * mean;
    var = fmaxf(var, 0.f);
    const float sc = bn_w[idx] * rsqrtf(var + 1e-5f);
    scale[idx] = sc;
    shift[idx] = bn_b[idx] - mean * sc;
}

// ---------------------------------------------------------------------------
// Kernel 3: recompute mm, a = relu(mm*scale + shift) -> same LDS buffer
// (wave-private rows, same-wave DS ops are in-order => no barrier), then
// y = a @ w2 + b2.
// ---------------------------------------------------------------------------
__global__ __launch_bounds__(NTHREADS)
void gemm2_kernel(const float* __restrict__ x,
                  const float* __restrict__ pw1,
                  const float* __restrict__ pw2,
                  const float* __restrict__ scale,
                  const float* __restrict__ shift,
                  const float* __restrict__ b2,
                  float* __restrict__ y,
                  int N)
{
    __shared__ float xs[ROWS * 128];             // 64 KB: x tile, then a tile

    const int head = blockIdx.y;
    const int row0 = blockIdx.x * ROWS;
    const int tid  = threadIdx.x;

    stage_x(x, xs, row0, head, N, tid);
    __syncthreads();

    const int wave = tid >> 5;
    const int lane = tid & 31;
    const int m    = lane & 15;
    const int koff = (lane < 16) ? 0 : 2;
    const int rA   = (wave << 4) + m;
    const int sA   = (rA & 15) << 2;

    const float* arow  = xs + rA * 128 + koff;
    const float* pw1hl = pw1 + (size_t)head * (KSTEPS * 8 * 64) + lane * 2;

    // ---- GEMM1 (recompute) ----
    v8f acc[8];
    #pragma unroll
    for (int t = 0; t < 8; ++t)
        #pragma unroll
        for (int j = 0; j < 8; ++j) acc[t][j] = 0.f;

    gemm1_core(arow, pw1hl, sA, acc);

    // ---- normalize + ReLU into this wave's own 16 LDS rows ----
    const int rhalf = (lane < 16) ? 0 : 8;
    #pragma unroll
    for (int t = 0; t < 8; ++t) {
        const int n = t * 16 + m;
        const float sc = scale[head * HID + n];
        const float sh = shift[head * HID + n];
        #pragma unroll
        for (int j = 0; j < 8; ++j) {
            const int r = (wave << 4) + rhalf + j;   // D-layout row
            xs[sw_idx(r, n)] = fmaxf(acc[t][j] * sc + sh, 0.f);
        }
    }

    // ---- GEMM2: a[16x128] @ w2[128x64] per wave ----
    const float* pw2hl = pw2 + (size_t)head * (KSTEPS * 4 * 64) + lane * 2;

    v8f acc2[4];
    #pragma unroll
    for (int t = 0; t < 4; ++t)
        #pragma unroll
        for (int j = 0; j < 8; ++j) acc2[t][j] = 0.f;

    #pragma unroll
    for (int k = 0; k < KSTEPS; ++k) {
        const v2f a = *(const v2f*)(arow + ((4 * k) ^ sA));
        #pragma unroll
        for (int t = 0; t < 4; ++t) {
            const v2f b = *(const v2f*)(pw2hl + (k * 4 + t) * 64);
            acc2[t] = wmma_f32(a, b, acc2[t]);
        }
    }

    // ---- + b2, store y (coalesced 64 B runs per lane half) ----
    const float* b2h = b2 + head * FOUT;
    #pragma unroll
    for (int t = 0; t < 4; ++t) {
        const int n = t * 16 + m;
        const float bias = b2h[n];
        #pragma unroll
        for (int j = 0; j < 8; ++j) {
            const int rg = row0 + (wave << 4) + rhalf + j;
            if (rg < N)
                y[(size_t)rg * (HEADS * FOUT) + head * FOUT + n] = acc2[t][j] + bias;
        }
    }
}

// ---------------------------------------------------------------------------
extern "C" void kernel_launch(void* const* d_in, const int* in_sizes, int n_in,
                              void* d_out, int out_size, void* d_ws, size_t ws_size,
                              hipStream_t stream) {
    const float* x    = (const float*)d_in[0];
    const float* w1   = (const float*)d_in[1];
    // d_in[2] = b1: cancels under training-mode BN -> unused.
    const float* bn_w = (const float*)d_in[3];
    const float* bn_b = (const float*)d_in[4];
    const float* w2   = (const float*)d_in[5];
    const float* b2   = (const float*)d_in[6];
    float* y = (float*)d_out;

    const int N = in_sizes[0] / (HEADS * FIN);   // 100000

    // Workspace layout (floats): 16 KB stats/coeffs + 768 KB packed weights.
    float* gsum   = (float*)d_ws;                        // [1024]
    float* gsumsq = gsum   + HEADS * HID;                // [1024]
    float* scale  = gsumsq + HEADS * HID;                // [1024]
    float* shift  = scale  + HEADS * HID;                // [1024]
    float* pw1    = shift  + HEADS * HID;                // [131072]
    float* pw2    = pw1    + HEADS * KSTEPS * 8 * 64;    // [65536]

    hipMemsetAsync(d_ws, 0, 2 * HEADS * HID * sizeof(float), stream);

    pack_weights_kernel<<<dim3((HEADS * KSTEPS * 8 * 64 + NTHREADS - 1) / NTHREADS),
                          NTHREADS, 0, stream>>>(w1, w2, pw1, pw2);

    dim3 grid((N + ROWS - 1) / ROWS, HEADS);
    gemm1_stats_kernel<<<grid, NTHREADS, 0, stream>>>(x, pw1, gsum, gsumsq, N);
    finalize_kernel<<<dim3(HEADS), dim3(128), 0, stream>>>(
        gsum, gsumsq, bn_w, bn_b, scale, shift, N);
    gemm2_kernel<<<grid, NTHREADS, 0, stream>>>(
        x, pw1, pw2, scale, shift, b2, y, N);
}